// DTWLayerPerChannel_14259291422991
// MI455X (gfx1250) — compile-verified
//
#include <hip/hip_runtime.h>
#include <cstdint>

// Problem constants (from reference): x (B,C,T) f32, patts (P,L) f32,
// out (B, P*C, L, LOUT) f32.
#define B_    32
#define C_    6
#define T_    512
#define P_    16
#define L_    24
#define LOUT_ 256
#define TJ    32                      // output columns staged per LDS flush
#define W_    0.9085176899f           // 0.1^(1/24)

typedef __attribute__((ext_vector_type(4))) unsigned int tdm_v4u_t;
typedef __attribute__((ext_vector_type(8))) int          tdm_v8i_t;
typedef __attribute__((ext_vector_type(4))) int          tdm_v4i_t;

__global__ __launch_bounds__(512)
void dtw_minplus_scan_kernel(const float* __restrict__ x,
                             const float* __restrict__ patts,
                             float* __restrict__ out) {
  __shared__ float lds_x[T_];                 // 2 KB: one x row per workgroup
  __shared__ float tile[P_][L_][TJ + 1];      // 50688 B: per-wave store-staging

  const int b    = blockIdx.x / C_;
  const int c    = blockIdx.x - b * C_;
  const int wave = threadIdx.x >> 5;          // == pattern index p (16 waves)
  const int lane = threadIdx.x & 31;          // == pattern position i (24 used)

  // ---- CDNA5 TDM: DMA x[b,c,0:512] (2 KB) global -> LDS, wave 0 only ----
  if (threadIdx.x < 32) {
    unsigned long long ga =
        (unsigned long long)(uintptr_t)(x + ((size_t)b * C_ + c) * T_);
    unsigned lds_off = (unsigned)(uintptr_t)(&lds_x[0]); // low 32b of flat LDS addr

    tdm_v4u_t g0;
    g0.x = 1u;                                           // count = 1 descriptor
    g0.y = lds_off;                                      // LDS byte address
    g0.z = (unsigned)(ga & 0xFFFFFFFFull);               // global_addr[31:0]
    g0.w = (unsigned)((ga >> 32) & 0x01FFFFFFull)        // global_addr[56:32]
           | 0x80000000u;                                // type = 2 ("image")

    tdm_v8i_t g1;
    g1[0] = 0x20000;                 // workgroup_mask=0, data_size=2 (4-byte)
    g1[1] = (int)(T_ << 16);         // tensor_dim0 = 512 (low half)
    g1[2] = (int)(1 << 16);          // tensor_dim0 hi=0, tensor_dim1 = 1
    g1[3] = (int)(T_ << 16);         // tensor_dim1 hi=0, tile_dim0 = 512
    g1[4] = 1;                       // tile_dim1 = 1, tile_dim2 = 0
    g1[5] = T_;                      // tensor_dim0_stride = 512 (low 32b)
    g1[6] = 0;                       // stride hi / tensor_dim1_stride lo
    g1[7] = 0;
    tdm_v4i_t gz4 = {0, 0, 0, 0};    // groups 2/3 unused (<=2D tensor)
    tdm_v8i_t gz8 = {0, 0, 0, 0, 0, 0, 0, 0};

    __builtin_amdgcn_tensor_load_to_lds(g0, g1, gz4, gz4, gz8, 0);
    __builtin_amdgcn_s_wait_tensorcnt(0);
  }
  __syncthreads();                   // LDS x row visible to all 16 waves

  const float w   = W_;
  const float BIG = 3.0e38f;         // (min,+) identity-ish; BIG+x -> inf, ok
  const float pi  = (lane < L_) ? patts[wave * L_ + lane] : 0.0f;

  // ---- column j = 0: DTW[i,0] = inclusive prefix-sum of d[i,0] over i ----
  float x0 = lds_x[0];
  float df = pi - x0;
  float prev = df * df;              // lane i holds DTW[i, j-1]
  #pragma unroll
  for (int dlt = 1; dlt < 32; dlt <<= 1) {
    float t = __shfl_up(prev, (unsigned)dlt);
    if (lane >= dlt) prev += t;
  }

  const size_t obase =
      (((size_t)b * (P_ * C_) + (size_t)wave * C_ + c) * L_) * (size_t)LOUT_;

  // ---- main scan over time: Kogge-Stone (min,+) scan across lanes ----
  for (int j = 1; j < T_; ++j) {
    float xj = lds_x[j];             // uniform LDS address -> broadcast read
    float d = pi - xj;  d *= d;      // d[i,j]

    float prev_up = __shfl_up(prev, 1u);   // DTW[i-1, j-1]
    float cc, aa;                    // v[i] = min(cc[i], aa[i] + v[i-1])
    if (lane == 0) { cc = d + w * prev;                 aa = BIG; }
    else           { cc = d + w * fminf(prev, prev_up); aa = d;   }

    #pragma unroll
    for (int dlt = 1; dlt < 32; dlt <<= 1) {   // 5-step min-plus scan
      float cs = __shfl_up(cc, (unsigned)dlt);
      float as = __shfl_up(aa, (unsigned)dlt);
      if (lane >= dlt) {
        cc = fminf(cc, aa + cs);     // compose affine (min,+) maps
        aa = aa + as;
      }
    }
    prev = cc;                       // DTW[i, j]

    if (j >= T_ - LOUT_) {
      int jo = j - (T_ - LOUT_);     // output column 0..255
      int jt = jo & (TJ - 1);
      if (lane < L_) tile[wave][lane][jt] = prev;   // stride-33: conflict-free
      if (jt == TJ - 1) {            // uniform across block (j is uniform)
        __syncthreads();
        int jb = jo - (TJ - 1);
        #pragma unroll 4
        for (int row = 0; row < L_; ++row) {
          // coalesced 128B row stores: lanes cover 32 consecutive columns
          out[obase + (size_t)row * LOUT_ + (size_t)(jb + lane)] =
              tile[wave][row][lane];
        }
        __syncthreads();
      }
    }
  }
}

extern "C" void kernel_launch(void* const* d_in, const int* in_sizes, int n_in,
                              void* d_out, int out_size, void* d_ws, size_t ws_size,
                              hipStream_t stream) {
  (void)in_sizes; (void)n_in; (void)out_size; (void)d_ws; (void)ws_size;
  const float* x     = (const float*)d_in[0];   // (32, 6, 512) f32
  const float* patts = (const float*)d_in[1];   // (16, 24) f32
  float* out         = (float*)d_out;           // (32, 96, 24, 256) f32

  dim3 grid(B_ * C_);                            // 192 workgroups, one per (b,c)
  dim3 block(512);                               // 16 waves = 16 patterns
  dtw_minplus_scan_kernel<<<grid, block, 0, stream>>>(x, patts, out);
}